// ModSTDP_58823872086838
// MI455X (gfx1250) — compile-verified
//
#include <hip/hip_runtime.h>
#include <cstdint>
#include <cstddef>

// ModSTDP update for MI455X (gfx1250).
// Memory-bound streaming kernel: 512 MiB spike stream + 64 MiB W read + 64 MiB W write
// => ~28.8 us floor at 23.3 TB/s. No matrix structure -> no WMMA; the CDNA5-specific
// path exercised is the async global->LDS data mover (ASYNCcnt) with double-buffered
// prefetch, plus NT temporal hints to keep the one-shot spike stream out of L2.

typedef int   v4i __attribute__((ext_vector_type(4)));
typedef float v4f __attribute__((ext_vector_type(4)));

// Address-space-qualified vector pointers for the async-to-LDS builtin
// (clang reports its params as pointers to 16-byte int vectors).
typedef __attribute__((address_space(1))) v4i gv4i;   // global
typedef __attribute__((address_space(3))) v4i lv4i;   // LDS

#define THREADS 128                  // 4 wave32 waves per block
#define VEC     4                    // b128 per thread
#define CHUNK   (THREADS * VEC)      // 512 synapses per chunk
#define CPB     8                    // chunks per block -> block spans 4096 synapses
#define TW      8                    // temporal window

#if __has_builtin(__builtin_amdgcn_global_load_async_to_lds_b128) && \
    __has_builtin(__builtin_amdgcn_s_wait_asynccnt)
#define USE_ASYNC 1
#else
#define USE_ASYNC 0
#endif

__device__ __forceinline__ uint32_t hash32(uint32_t x) {
  // lowbias32 finisher: 2 muls, good avalanche, deterministic per element.
  x ^= x >> 16; x *= 0x7feb352du;
  x ^= x >> 15; x *= 0x846ca68bu;
  x ^= x >> 16;
  return x;
}

__global__ __launch_bounds__(THREADS)
void modstdp_kernel(const int* __restrict__ in_spk,   // [8, N, S] int32 0/1
                    const int* __restrict__ out_spk,  // [8, N]    int32 0/1
                    const float* __restrict__ W,      // [N, S]
                    float* __restrict__ Wout,         // [N, S]
                    int N, int S)
{
  const int    n     = blockIdx.y;
  const int    tid   = threadIdx.x;
  const size_t row   = (size_t)n * (size_t)S;
  const size_t plane = (size_t)N * (size_t)S;
  const size_t sBase = (size_t)blockIdx.x * (size_t)(CHUNK * CPB);

  // Per-neuron output spike time: block-uniform address -> scalar loads (KMcnt).
  int osum = 0;
#pragma unroll
  for (int t = 0; t < TW; ++t) osum += out_spk[t * N + n];
  const int  out_t     = TW - osum;
  const bool out_fired = out_t < TW;

#if USE_ASYNC
  __shared__ int lbuf[2][TW][CHUNK];   // 32 KB: double-buffered spike staging
  auto issue = [&](int c, int b) {
    const size_t s0 = sBase + (size_t)c * CHUNK + (size_t)tid * VEC;
#pragma unroll
    for (int t = 0; t < TW; ++t) {
      __builtin_amdgcn_global_load_async_to_lds_b128(
          (gv4i*)(in_spk + plane * (size_t)t + row + s0),
          (lv4i*)(&lbuf[b][t][tid * VEC]),
          0, 0);
    }
  };
  issue(0, 0);
#endif

  for (int c = 0; c < CPB; ++c) {
    const size_t s0 = sBase + (size_t)c * CHUNK + (size_t)tid * VEC;
    int sum0 = 0, sum1 = 0, sum2 = 0, sum3 = 0;

#if USE_ASYNC
    const int b = c & 1;
    if (c + 1 < CPB) {
      issue(c + 1, b ^ 1);                      // prefetch next chunk
      __builtin_amdgcn_s_wait_asynccnt(8);      // current chunk's 8 loads landed
    } else {
      __builtin_amdgcn_s_wait_asynccnt(0);
    }
    asm volatile("" ::: "memory");              // keep LDS reads below the wait
#pragma unroll
    for (int t = 0; t < TW; ++t) {
      v4i v = *(const v4i*)&lbuf[b][t][tid * VEC];   // ds_load_b128
      sum0 += v.x; sum1 += v.y; sum2 += v.z; sum3 += v.w;
    }
#else
#pragma unroll
    for (int t = 0; t < TW; ++t) {
      v4i v = __builtin_nontemporal_load(
          (const v4i*)(in_spk + plane * (size_t)t + row + s0));
      sum0 += v.x; sum1 += v.y; sum2 += v.z; sum3 += v.w;
    }
#endif

    v4f wv = __builtin_nontemporal_load((const v4f*)(W + row + s0));
    const int sums[VEC] = {sum0, sum1, sum2, sum3};
    v4f res;
#pragma unroll
    for (int j = 0; j < VEC; ++j) {
      float w = fminf(fmaxf(wv[j], 0.0f), 8.0f);
      const int  in_t     = TW - sums[j];
      const bool in_fired = in_t < TW;

      // 59 random bits per element from two 32-bit hashes.
      const uint32_t idx = (uint32_t)(row + s0) + (uint32_t)j;
      const uint32_t r1  = hash32(idx ^ 0x9e3779b9u);
      const uint32_t r2  = hash32(idx + 0x7f4a7c15u);

      const float wn     = w * 0.125f;
      const bool  fplus  = (float)( r1        & 0xFFFu) * (1.0f/4096.0f) < wn * (2.0f - wn);
      const bool  fminus = (float)((r1 >> 12) & 0xFFFu) * (1.0f/4096.0f) < 1.0f - wn * wn;
      const bool  umin   = ((r1 >> 24) & 0x7Fu) <  4u;   // 4/128
      const bool  ucap   = ( r2        & 0x7Fu) < 10u;   // 10/128
      const bool  uminb  = ((r2 >>  7) & 0x7Fu) < 10u;   // 10/128
      const bool  usrch  = ((r2 >> 14) & 0x7Fu) <  1u;   // 1/128
      const bool  uback  = ((r2 >> 21) & 0x7Fu) < 96u;   // 96/128

      const bool mfp = fplus  | umin;   // max(fplus,  umin) over {0,1}
      const bool mfm = fminus | umin;   // max(fminus, umin)

      // Branches are mutually exclusive; exactly one delta applies.
      float d = 0.0f;
      if (in_fired & out_fired) {
        d = (in_t <= out_t) ? ((ucap  & mfp) ?  1.0f : 0.0f)
                            : ((uminb & mfm) ? -1.0f : 0.0f);
      } else if (in_fired) {            // output silent (out_t == 8)
        d = (usrch & mfp) ?  1.0f : 0.0f;
      } else if (out_fired) {           // input silent (in_t == 8)
        d = (uback & mfm) ? -1.0f : 0.0f;
      }
      res[j] = w + d;
    }
    __builtin_nontemporal_store(res, (v4f*)(Wout + row + s0));
  }
}

extern "C" void kernel_launch(void* const* d_in, const int* in_sizes, int n_in,
                              void* d_out, int out_size, void* d_ws, size_t ws_size,
                              hipStream_t stream) {
  (void)n_in; (void)out_size; (void)d_ws; (void)ws_size;
  const int*   in_spk  = (const int*)d_in[0];   // [8, N, S]
  const int*   out_spk = (const int*)d_in[1];   // [8, N]
  const float* W       = (const float*)d_in[2]; // [N, S]
  float*       Wout    = (float*)d_out;

  const int N = in_sizes[1] / TW;               // (T*N)/T
  const int S = (N > 0) ? (in_sizes[2] / N) : 0;

  const int span = CHUNK * CPB;                 // 4096 synapses per block
  dim3 grid((unsigned)((S + span - 1) / span), (unsigned)N, 1);
  dim3 block(THREADS, 1, 1);
  modstdp_kernel<<<grid, block, 0, stream>>>(in_spk, out_spk, W, Wout, N, S);
}